// GAT_63058709840259
// MI455X (gfx1250) — compile-verified
//
#include <hip/hip_runtime.h>
#include <hip/hip_bf16.h>
#include <math.h>

// ---------------------------------------------------------------------------
// GAT (2-layer, PyG-style) for MI455X / gfx1250.
//   Layer1: h1 = x@W1 -> edge softmax -> out1 (+b1, ELU)
//   Layer2: h2 = out1@W2 -> edge softmax -> d_out (+b2)
// GEMMs: V_WMMA_F32_16X16X4_F32 (exact fp32) with the B strip staged in LDS
// via GLOBAL_LOAD_ASYNC_TO_LDS_B128 (ASYNCcnt) when the builtin is available.
// ---------------------------------------------------------------------------

typedef __attribute__((ext_vector_type(2))) float v2f;
typedef __attribute__((ext_vector_type(8))) float v8f;
typedef int v4i __attribute__((vector_size(16)));
typedef __attribute__((address_space(1))) v4i* gptr_v4i;
typedef __attribute__((address_space(3))) v4i* lptr_v4i;

#define NEG_SLOPE 0.2f

#if __has_builtin(__builtin_amdgcn_global_load_async_to_lds_b128)
#define HAVE_ASYNC_LDS 1
#endif

// --------------------------- WMMA f32 GEMM ---------------------------------
// Block = 8 waves; each block computes a 128(M) x 64(N) tile of C = A x B.
// The B[K, 64] column strip is staged in LDS once per block (all 8 waves
// share it). f32 WMMA operand layouts (ISA 7.12.2):
//   A 16x4 : lane(0..15)=row M, VGPR j -> K = 2*(lane>>4)+j
//   B 4x16 : lane&15 = col N,   VGPR j -> K = 2*(lane>>4)+j
//   C 16x16: VGPR r -> M = r + 8*(lane>>4), N = lane&15
__global__ __launch_bounds__(256) void wmma_gemm_f32(
    const float* __restrict__ A, const float* __restrict__ B,
    float* __restrict__ C, int M, int K, int Ncol) {
  extern __shared__ float Bs[];                    // K*64 floats (<=128KB)
  const int tid = threadIdx.x;
  const int colBase = blockIdx.y * 64;

  // ---- stage B[0..K-1][colBase..colBase+63] into LDS, 16B per lane ----
  const int quads = K * 16;                        // number of 16B chunks
  for (int q = tid; q < quads; q += 256) {
    const int k  = q >> 4;
    const int c4 = (q & 15) * 4;
    const float* gsrc = B + (size_t)k * Ncol + colBase + c4;
    float*       ldst = Bs + k * 64 + c4;
#if defined(HAVE_ASYNC_LDS)
    __builtin_amdgcn_global_load_async_to_lds_b128(
        (gptr_v4i)(void*)(float*)gsrc, (lptr_v4i)(void*)ldst, 0, 0);
#else
    *(float4*)ldst = *(const float4*)gsrc;
#endif
  }
#if defined(HAVE_ASYNC_LDS)
#if __has_builtin(__builtin_amdgcn_s_wait_asynccnt)
  __builtin_amdgcn_s_wait_asynccnt(0);
#else
  asm volatile("s_wait_asynccnt 0" ::: "memory");
#endif
#endif
  __syncthreads();

  const int wave = tid >> 5;
  const int lane = tid & 31;
  const int rowTile = blockIdx.x * 8 + wave;
  const int row0 = rowTile * 16;
  if (row0 >= M) return;                           // wave-uniform, after barrier
  const int half = lane >> 4;
  const int l15  = lane & 15;
  int aRow = row0 + l15;
  if (aRow >= M) aRow = M - 1;                     // clamp reads; stores guarded

  const float* Arow = A + (size_t)aRow * K;
  v8f c0 = {}, c1 = {}, c2 = {}, c3 = {};

  for (int k = 0; k < K; k += 4) {
    const int ka = k + half * 2;                   // ka even -> 8B-aligned A load
    const v2f a = *(const v2f*)(Arow + ka);
    const float* Bk = Bs + ka * 64 + l15;          // LDS reads (ds_load)
    const v2f b0 = {Bk[0],  Bk[64]};
    const v2f b1 = {Bk[16], Bk[80]};
    const v2f b2 = {Bk[32], Bk[96]};
    const v2f b3 = {Bk[48], Bk[112]};
    c0 = __builtin_amdgcn_wmma_f32_16x16x4_f32(false, a, false, b0, (short)0, c0, false, false);
    c1 = __builtin_amdgcn_wmma_f32_16x16x4_f32(false, a, false, b1, (short)0, c1, false, false);
    c2 = __builtin_amdgcn_wmma_f32_16x16x4_f32(false, a, false, b2, (short)0, c2, false, false);
    c3 = __builtin_amdgcn_wmma_f32_16x16x4_f32(false, a, false, b3, (short)0, c3, false, false);
  }

  for (int r = 0; r < 8; ++r) {
    const int m = row0 + r + 8 * half;
    if (m < M) {
      float* Crow = C + (size_t)m * Ncol + colBase + l15;
      Crow[0]  = c0[r];
      Crow[16] = c1[r];
      Crow[32] = c2[r];
      Crow[48] = c3[r];
    }
  }
}

// ------------------------ attention score dots -----------------------------
__global__ __launch_bounds__(256) void scores_kernel(
    const float* __restrict__ h, const float* __restrict__ a_src,
    const float* __restrict__ a_dst, float* __restrict__ es,
    float* __restrict__ ed, int Nn, int H, int C) {
  const int wid  = (int)((blockIdx.x * blockDim.x + threadIdx.x) >> 5);
  const int lane = threadIdx.x & 31;
  if (wid >= Nn * H) return;
  const int n  = wid / H;
  const int hh = wid - n * H;
  const float* hv = h + (size_t)n * H * C + (size_t)hh * C;
  const float* as = a_src + (size_t)hh * C;
  const float* ad = a_dst + (size_t)hh * C;
  float s = 0.f, d = 0.f;
  for (int c = lane; c < C; c += 32) {
    const float v = hv[c];
    s += v * as[c];
    d += v * ad[c];
  }
  for (int off = 16; off > 0; off >>= 1) {
    s += __shfl_xor(s, off, 32);
    d += __shfl_xor(d, off, 32);
  }
  if (lane == 0) { es[wid] = s; ed[wid] = d; }
}

// ----------------------------- utilities -----------------------------------
__global__ void fill_kernel(float* p, float v, size_t n) {
  size_t i = (size_t)blockIdx.x * blockDim.x + threadIdx.x;
  if (i < n) p[i] = v;
}

__global__ void init_out_bias_kernel(float* out, const float* b, int Nn, int F) {
  size_t i = (size_t)blockIdx.x * blockDim.x + threadIdx.x;
  if (i < (size_t)Nn * F) out[i] = b[i % F];
}

__device__ __forceinline__ float atomicMaxFloat(float* addr, float v) {
  if (v >= 0.f)
    return __int_as_float(atomicMax((int*)addr, __float_as_int(v)));
  else
    return __uint_as_float(atomicMin((unsigned int*)addr, __float_as_uint(v)));
}

// --------------------------- edge phase ------------------------------------
__global__ __launch_bounds__(256) void edge_logits_max_kernel(
    const long long* __restrict__ srcIdx, const long long* __restrict__ dstIdx,
    const float* __restrict__ es, const float* __restrict__ ed,
    float* __restrict__ eBuf, float* __restrict__ m, int E, int Etot, int H) {
  const int e = blockIdx.x * blockDim.x + threadIdx.x;
  if (e >= Etot) return;
  const int s = (e < E) ? (int)srcIdx[e] : (e - E);
  const int d = (e < E) ? (int)dstIdx[e] : (e - E);
  for (int h = 0; h < H; ++h) {
    float v = es[s * H + h] + ed[d * H + h];
    v = (v > 0.f) ? v : NEG_SLOPE * v;
    eBuf[(size_t)e * H + h] = v;
    atomicMaxFloat(&m[d * H + h], v);
  }
}

__global__ __launch_bounds__(256) void edge_exp_sum_kernel(
    const long long* __restrict__ dstIdx, float* __restrict__ eBuf,
    const float* __restrict__ m, float* __restrict__ den, int E, int Etot, int H) {
  const int e = blockIdx.x * blockDim.x + threadIdx.x;
  if (e >= Etot) return;
  const int d = (e < E) ? (int)dstIdx[e] : (e - E);
  for (int h = 0; h < H; ++h) {
    const float p = expf(eBuf[(size_t)e * H + h] - m[d * H + h]);
    eBuf[(size_t)e * H + h] = p;
    atomicAdd(&den[d * H + h], p);
  }
}

// out[dst] += h[src] * alpha.  One wave per edge; float2 gathers (b64).
__global__ __launch_bounds__(256) void edge_message_kernel(
    const long long* __restrict__ srcIdx, const long long* __restrict__ dstIdx,
    const float* __restrict__ h, const float* __restrict__ p,
    const float* __restrict__ den, float* __restrict__ out,
    int E, int Etot, int H, int logC) {
  const int wid  = (int)((blockIdx.x * blockDim.x + threadIdx.x) >> 5);
  const int lane = threadIdx.x & 31;
  if (wid >= Etot) return;
  const int s = (wid < E) ? (int)srcIdx[wid] : (wid - E);
  const int d = (wid < E) ? (int)dstIdx[wid] : (wid - E);
  const int HC = H << logC;
  const float* hs = h + (size_t)s * HC;
  float* od = out + (size_t)d * HC;
  for (int pr = lane; pr < (HC >> 1); pr += 32) {
    const int ch = pr * 2;
    const int hh = ch >> logC;                     // both channels in same head
    const float alpha = p[(size_t)wid * H + hh] / (den[d * H + hh] + 1e-16f);
    const float2 hv = *(const float2*)(hs + ch);
    atomicAdd(&od[ch],     hv.x * alpha);
    atomicAdd(&od[ch + 1], hv.y * alpha);
  }
}

__global__ void bias_elu_kernel(float* __restrict__ x, const float* __restrict__ b,
                                int Nn, int F) {
  size_t i = (size_t)blockIdx.x * blockDim.x + threadIdx.x;
  if (i >= (size_t)Nn * F) return;
  const float v = x[i] + b[i % F];
  x[i] = (v > 0.f) ? v : (expf(v) - 1.f);
}

// ---------------------------------------------------------------------------
extern "C" void kernel_launch(void* const* d_in, const int* in_sizes, int n_in,
                              void* d_out, int out_size, void* d_ws, size_t ws_size,
                              hipStream_t stream) {
  const float*     x      = (const float*)d_in[0];
  const long long* eidx   = (const long long*)d_in[1];   // int64 in reference
  const float*     W1     = (const float*)d_in[2];
  const float*     a_src1 = (const float*)d_in[3];
  const float*     a_dst1 = (const float*)d_in[4];
  const float*     b1     = (const float*)d_in[5];
  const float*     W2     = (const float*)d_in[6];
  const float*     a_src2 = (const float*)d_in[7];
  const float*     a_dst2 = (const float*)d_in[8];
  const float*     b2     = (const float*)d_in[9];

  const int IN_CH = 256, HID = 128, HEADS = 4, OUT_CH = 64;
  const int F1 = HEADS * HID;                 // 512
  const int N  = in_sizes[0] / IN_CH;         // 25000
  const int E  = in_sizes[1] / 2;             // 400000
  const int Etot = E + N;                     // self-loops appended
  const long long* srcIdx = eidx;
  const long long* dstIdx = eidx + E;

  // ---- workspace carve-up (floats) ----
  float* w = (float*)d_ws;
  size_t off = 0;
  float* h1   = w + off; off += (size_t)N * F1;       // 12.8M
  float* out1 = w + off; off += (size_t)N * F1;       // 12.8M
  float* es1  = w + off; off += (size_t)N * HEADS;
  float* ed1  = w + off; off += (size_t)N * HEADS;
  float* m1   = w + off; off += (size_t)N * HEADS;
  float* den1 = w + off; off += (size_t)N * HEADS;
  float* e1b  = w + off; off += (size_t)Etot * HEADS; // 1.7M
  float* h2   = w + off; off += (size_t)N * OUT_CH;   // 1.6M
  float* es2  = w + off; off += (size_t)N;
  float* ed2  = w + off; off += (size_t)N;
  float* m2   = w + off; off += (size_t)N;
  float* den2 = w + off; off += (size_t)N;
  float* e2b  = w + off; off += (size_t)Etot;
  (void)ws_size; (void)n_in; (void)out_size;

  const int BLK = 256;
  const int rowTiles = (N + 15) / 16;
  const dim3 gemmGrid1((rowTiles + 7) / 8, F1 / 64);
  const dim3 gemmGrid2((rowTiles + 7) / 8, OUT_CH / 64);
  const size_t shmem1 = (size_t)IN_CH * 64 * sizeof(float);  // 64 KB
  const size_t shmem2 = (size_t)F1 * 64 * sizeof(float);     // 128 KB (WGP has 320 KB)
  const int edgeBlocks = (Etot + BLK - 1) / BLK;
  const int edgeWaveBlocks = (Etot + 7) / 8;          // 8 waves (edges) per block

  // ================= Layer 1 =================
  wmma_gemm_f32<<<gemmGrid1, BLK, shmem1, stream>>>(x, W1, h1, N, IN_CH, F1);
  scores_kernel<<<(N * HEADS + 7) / 8, BLK, 0, stream>>>(h1, a_src1, a_dst1, es1, ed1,
                                                         N, HEADS, HID);
  fill_kernel<<<((size_t)N * HEADS + BLK - 1) / BLK, BLK, 0, stream>>>(m1, -INFINITY, (size_t)N * HEADS);
  fill_kernel<<<((size_t)N * HEADS + BLK - 1) / BLK, BLK, 0, stream>>>(den1, 0.f, (size_t)N * HEADS);
  fill_kernel<<<((size_t)N * F1 + BLK - 1) / BLK, BLK, 0, stream>>>(out1, 0.f, (size_t)N * F1);

  edge_logits_max_kernel<<<edgeBlocks, BLK, 0, stream>>>(srcIdx, dstIdx, es1, ed1, e1b, m1,
                                                         E, Etot, HEADS);
  edge_exp_sum_kernel<<<edgeBlocks, BLK, 0, stream>>>(dstIdx, e1b, m1, den1, E, Etot, HEADS);
  edge_message_kernel<<<edgeWaveBlocks, BLK, 0, stream>>>(srcIdx, dstIdx, h1, e1b, den1, out1,
                                                          E, Etot, HEADS, 7 /*log2(128)*/);
  bias_elu_kernel<<<((size_t)N * F1 + BLK - 1) / BLK, BLK, 0, stream>>>(out1, b1, N, F1);

  // ================= Layer 2 =================
  wmma_gemm_f32<<<gemmGrid2, BLK, shmem2, stream>>>(out1, W2, h2, N, F1, OUT_CH);
  scores_kernel<<<(N + 7) / 8, BLK, 0, stream>>>(h2, a_src2, a_dst2, es2, ed2, N, 1, OUT_CH);
  fill_kernel<<<((size_t)N + BLK - 1) / BLK, BLK, 0, stream>>>(m2, -INFINITY, (size_t)N);
  fill_kernel<<<((size_t)N + BLK - 1) / BLK, BLK, 0, stream>>>(den2, 0.f, (size_t)N);
  // H=1 -> mean over heads is identity; pre-load bias, scatter-add messages on top.
  init_out_bias_kernel<<<((size_t)N * OUT_CH + BLK - 1) / BLK, BLK, 0, stream>>>(
      (float*)d_out, b2, N, OUT_CH);

  edge_logits_max_kernel<<<edgeBlocks, BLK, 0, stream>>>(srcIdx, dstIdx, es2, ed2, e2b, m2,
                                                         E, Etot, 1);
  edge_exp_sum_kernel<<<edgeBlocks, BLK, 0, stream>>>(dstIdx, e2b, m2, den2, E, Etot, 1);
  edge_message_kernel<<<edgeWaveBlocks, BLK, 0, stream>>>(srcIdx, dstIdx, h2, e2b, den2,
                                                          (float*)d_out, E, Etot, 1,
                                                          6 /*log2(64)*/);
}